// WeightedAverageWirelength_67619965108845
// MI455X (gfx1250) — compile-verified
//
#include <hip/hip_runtime.h>
#include <hip/hip_bf16.h>

// Weighted-average wirelength (WAWL) for MI455X / gfx1250.
// Segmented scatter-reduction workload: no matrix structure -> no WMMA.
// Memory-system optimized: b128 coalesced streams, packed b64 min/max
// gathers, integer-trick float atomic min/max, f32 atomic segment sums
// (L2-resident arrays), wave32 shuffle reductions, f64 atomic final
// accumulate, global_prefetch_b8 on the streamed arrays.

#define THREADS 256
#define PPT 4  // pins per thread

// ---------------------------------------------------------------------------
// Init per-net arrays. Coords are >= 0, so float bit patterns order as uints:
// max init = 0x00000000 (0.0f), min init = 0x7f800000 (+inf).
// mm is interleaved: mm[2*net] = max bits, mm[2*net+1] = min bits.
// ---------------------------------------------------------------------------
__global__ void wawl_init(unsigned* __restrict__ mm,
                          float* __restrict__ sums,
                          int num_nets,
                          double* acc, float* out) {
  int i = blockIdx.x * blockDim.x + threadIdx.x;
  if (i < num_nets) {
    mm[2 * (size_t)i]     = 0u;           // running max (bits)
    mm[2 * (size_t)i + 1] = 0x7f800000u;  // running min (bits) = +inf
    sums[i]                      = 0.0f;  // sum e^+
    sums[i + (size_t)num_nets]   = 0.0f;  // sum e^-
    sums[i + 2*(size_t)num_nets] = 0.0f;  // sum x*e^+
    sums[i + 3*(size_t)num_nets] = 0.0f;  // sum x*e^-
  }
  if (acc != nullptr && i == 0) { *acc = 0.0; *out = 0.0f; }
}

// ---------------------------------------------------------------------------
// Pass 1: per-net max/min via u32 atomics (valid for non-negative floats).
// ---------------------------------------------------------------------------
__device__ __forceinline__ void minmax_one(float c, int net,
                                           unsigned* __restrict__ mm) {
  unsigned b = __float_as_uint(c);
  atomicMax(mm + 2 * (size_t)net, b);      // global_atomic_max_u32
  atomicMin(mm + 2 * (size_t)net + 1, b);  // global_atomic_min_u32
}

__global__ void wawl_minmax(const float* __restrict__ coord,
                            const int* __restrict__ p2n,
                            unsigned* __restrict__ mm,
                            int num_pins) {
  int base = (blockIdx.x * blockDim.x + threadIdx.x) * PPT;
  __builtin_prefetch(coord + base + 8192, 0, 0);  // global_prefetch_b8
  __builtin_prefetch(p2n   + base + 8192, 0, 0);
  bool aligned = (((size_t)coord | (size_t)p2n) & 15u) == 0;
  if (aligned && base + PPT <= num_pins) {
    float4 c4 = *reinterpret_cast<const float4*>(coord + base);
    int4   n4 = *reinterpret_cast<const int4*>(p2n + base);
    minmax_one(c4.x, n4.x, mm);
    minmax_one(c4.y, n4.y, mm);
    minmax_one(c4.z, n4.z, mm);
    minmax_one(c4.w, n4.w, mm);
  } else {
    for (int k = 0; k < PPT; ++k) {
      int i = base + k;
      if (i < num_pins) minmax_one(coord[i], p2n[i], mm);
    }
  }
}

// ---------------------------------------------------------------------------
// Pass 2: one b64 gather of {max,min} per pin, stabilized exponentials,
// four f32 atomic segment sums per pin. exp args are <= 0 so results lie in
// (0,1]; the min-pin contributes exp(0)=1, so sum_n >= 1 for non-empty nets.
// ---------------------------------------------------------------------------
__device__ __forceinline__ void expsum_one(float c, int net, float ig,
                                           const uint2* __restrict__ mm,
                                           float* __restrict__ sp,
                                           float* __restrict__ sn,
                                           float* __restrict__ sxp,
                                           float* __restrict__ sxn) {
  uint2 b = mm[net];                        // single global_load_b64 gather
  float mx = __uint_as_float(b.x);
  float mn = __uint_as_float(b.y);
  float ep = __expf((c - mx) * ig);         // v_exp_f32
  float en = __expf((mn - c) * ig);
  atomicAdd(sp  + net, ep);                 // global_atomic_add_f32
  atomicAdd(sn  + net, en);
  atomicAdd(sxp + net, c * ep);
  atomicAdd(sxn + net, c * en);
}

__global__ void wawl_expsum(const float* __restrict__ coord,
                            const int* __restrict__ p2n,
                            const uint2* __restrict__ mm,
                            float* __restrict__ sums,
                            const float* __restrict__ inv_gamma,
                            int num_pins, int num_nets) {
  const float ig = inv_gamma[0];            // uniform -> scalar load
  float* sp  = sums;
  float* sn  = sums + (size_t)num_nets;
  float* sxp = sums + 2*(size_t)num_nets;
  float* sxn = sums + 3*(size_t)num_nets;
  int base = (blockIdx.x * blockDim.x + threadIdx.x) * PPT;
  __builtin_prefetch(coord + base + 8192, 0, 0);
  __builtin_prefetch(p2n   + base + 8192, 0, 0);
  bool aligned = (((size_t)coord | (size_t)p2n) & 15u) == 0;
  if (aligned && base + PPT <= num_pins) {
    float4 c4 = *reinterpret_cast<const float4*>(coord + base);
    int4   n4 = *reinterpret_cast<const int4*>(p2n + base);
    expsum_one(c4.x, n4.x, ig, mm, sp, sn, sxp, sxn);
    expsum_one(c4.y, n4.y, ig, mm, sp, sn, sxp, sxn);
    expsum_one(c4.z, n4.z, ig, mm, sp, sn, sxp, sxn);
    expsum_one(c4.w, n4.w, ig, mm, sp, sn, sxp, sxn);
  } else {
    for (int k = 0; k < PPT; ++k) {
      int i = base + k;
      if (i < num_pins)
        expsum_one(coord[i], p2n[i], ig, mm, sp, sn, sxp, sxn);
    }
  }
}

// ---------------------------------------------------------------------------
// Pass 3: per-net WL, masked & weighted, block-reduced (wave32 shuffles +
// LDS), one f64 atomic add per block into the accumulator.
// ---------------------------------------------------------------------------
__global__ void wawl_reduce(const float* __restrict__ sums,
                            const float* __restrict__ weights,
                            const unsigned char* __restrict__ mask,
                            int num_nets,
                            double* __restrict__ acc) {
  int n = blockIdx.x * blockDim.x + threadIdx.x;
  float v = 0.0f;
  if (n < num_nets) {
    float sp = sums[n];
    if (sp > 0.0f) {                  // guard empty nets
      float sn  = sums[n + (size_t)num_nets];
      float sxp = sums[n + 2*(size_t)num_nets];
      float sxn = sums[n + 3*(size_t)num_nets];
      float wl  = sxp / sp - sxn / sn;
      if (mask[n]) v = weights[n] * wl;
    }
  }
  // wave32 reduction
  #pragma unroll
  for (int o = 16; o > 0; o >>= 1) v += __shfl_down(v, o, 32);
  __shared__ float smem[THREADS / 32];
  int lane = threadIdx.x & 31;
  int wave = threadIdx.x >> 5;
  if (lane == 0) smem[wave] = v;
  __syncthreads();
  if (wave == 0) {
    v = (lane < THREADS / 32) ? smem[lane] : 0.0f;
    #pragma unroll
    for (int o = THREADS / 64; o > 0; o >>= 1) v += __shfl_down(v, o, 32);
    if (lane == 0) atomicAdd(acc, (double)v);   // global_atomic_add_f64
  }
}

__global__ void wawl_finalize(const double* __restrict__ acc,
                              float* __restrict__ out) {
  if (blockIdx.x == 0 && threadIdx.x == 0) out[0] = (float)(*acc);
}

// ---------------------------------------------------------------------------
// Launch. Inputs (setup_inputs order):
//   0 pos(2*NP f32) 1 pin2net(NP i32) 2 net_weights(NN f32, y-axis)
//   3 net_weights_x(NN f32) 4 net_mask(NN bool8) 5 pin_mask(unused)
//   6 inv_gamma(1 f32) 7 num_nets(scalar, ignored: use in_sizes)
// Workspace: mm (2*NN u32, interleaved max/min) + 4 sum arrays (4*NN f32)
//          + 1 double = 24*NN + 8 bytes (48 MB for NN = 2M).
// ---------------------------------------------------------------------------
extern "C" void kernel_launch(void* const* d_in, const int* in_sizes, int n_in,
                              void* d_out, int out_size, void* d_ws, size_t ws_size,
                              hipStream_t stream) {
  const float* pos          = (const float*)d_in[0];
  const int* p2n            = (const int*)d_in[1];
  const float* w_y          = (const float*)d_in[2];  // net_weights (y)
  const float* w_x          = (const float*)d_in[3];  // net_weights_x
  const unsigned char* mask = (const unsigned char*)d_in[4];
  const float* inv_gamma    = (const float*)d_in[6];

  const int num_pins = in_sizes[1];
  const int num_nets = in_sizes[2];

  char* w = (char*)d_ws;
  unsigned* mm  = (unsigned*)w;                       // 2*NN u32
  float* sums   = (float*)(mm + 2 * (size_t)num_nets);
  size_t acc_off = ((size_t)num_nets * 24u + 7u) & ~(size_t)7u;
  double* acc   = (double*)(w + acc_off);

  const int netBlocks  = (num_nets + THREADS - 1) / THREADS;
  const int pinThreads = (num_pins + PPT - 1) / PPT;
  const int pinBlocks  = (pinThreads + THREADS - 1) / THREADS;

  for (int axis = 0; axis < 2; ++axis) {
    const float* coord = pos + (size_t)axis * num_pins;
    const float* wgt   = (axis == 0) ? w_x : w_y;
    wawl_init<<<netBlocks, THREADS, 0, stream>>>(
        mm, sums, num_nets, (axis == 0) ? acc : nullptr, (float*)d_out);
    wawl_minmax<<<pinBlocks, THREADS, 0, stream>>>(
        coord, p2n, mm, num_pins);
    wawl_expsum<<<pinBlocks, THREADS, 0, stream>>>(
        coord, p2n, (const uint2*)mm, sums, inv_gamma, num_pins, num_nets);
    wawl_reduce<<<netBlocks, THREADS, 0, stream>>>(
        sums, wgt, mask, num_nets, acc);
  }
  wawl_finalize<<<1, 32, 0, stream>>>(acc, (float*)d_out);
}